// TransitionDown_22909355556977
// MI455X (gfx1250) — compile-verified
//
#include <hip/hip_runtime.h>

typedef __attribute__((ext_vector_type(2))) float v2f;
typedef __attribute__((ext_vector_type(8))) float v8f;

#define BB_   16
#define NN_   4096
#define CIN_  64
#define COUT_ 128
#define MM_   1024
#define KNB_  16
#define K1_   67
#define K1P_  68
#define CNT_  (BB_*MM_*KNB_)   /* 262144 */
#define EPS_  1e-5f
#define GROUPS_ 8
#define NBLK_ ((BB_*MM_)/GROUPS_)   /* 2048 */

// ---------------------------------------------------------------------------
// Farthest point sampling: one block per batch, 256 threads, 16 points/thread.
// ---------------------------------------------------------------------------
__global__ __launch_bounds__(256) void fps_kernel(const float* __restrict__ p1,
                                                  int* __restrict__ idx_out) {
    const int b = blockIdx.x, tid = threadIdx.x;
    const float* pb = p1 + b * NN_ * 3;
    float px[16], py[16], pz[16], dist[16];
#pragma unroll
    for (int i = 0; i < 16; ++i) {
        int pt = tid * 16 + i;
        px[i] = pb[pt * 3 + 0];
        py[i] = pb[pt * 3 + 1];
        pz[i] = pb[pt * 3 + 2];
        dist[i] = 1e10f;
    }
    __shared__ float s_red[256];
    __shared__ int   s_idx[256];
    __shared__ float s_c[3];
    int far = 0;
    for (int it = 0; it < MM_; ++it) {
        if (tid == 0) {
            idx_out[b * MM_ + it] = far;
            s_c[0] = pb[far * 3 + 0];
            s_c[1] = pb[far * 3 + 1];
            s_c[2] = pb[far * 3 + 2];
        }
        __syncthreads();
        float cx = s_c[0], cy = s_c[1], cz = s_c[2];
        float bv = -1.0f; int bi = 0;
#pragma unroll
        for (int i = 0; i < 16; ++i) {
            float dx = px[i] - cx, dy = py[i] - cy, dz = pz[i] - cz;
            float d  = dx * dx + dy * dy + dz * dz;
            float nd = fminf(dist[i], d);
            dist[i] = nd;
            if (nd > bv) { bv = nd; bi = tid * 16 + i; }   // '>' keeps first index
        }
        s_red[tid] = bv; s_idx[tid] = bi;
        __syncthreads();
        for (int s = 128; s > 0; s >>= 1) {
            if (tid < s && s_red[tid + s] > s_red[tid]) {
                s_red[tid] = s_red[tid + s];
                s_idx[tid] = s_idx[tid + s];
            }
            __syncthreads();
        }
        far = s_idx[0];
        __syncthreads();
    }
}

// ---------------------------------------------------------------------------
// Gather sampled centers p2 = p1[idx]; write both to ws copy and to d_out.
// ---------------------------------------------------------------------------
__global__ void gather_p2_kernel(const float* __restrict__ p1,
                                 const int* __restrict__ idx_fps,
                                 float* __restrict__ p2_ws,
                                 float* __restrict__ p2_out) {
    int t = blockIdx.x * blockDim.x + threadIdx.x;
    if (t >= BB_ * MM_) return;
    int b = t / MM_;
    int j = idx_fps[t];
    const float* src = p1 + (b * NN_ + j) * 3;
    float x = src[0], y = src[1], z = src[2];
    p2_ws[t * 3 + 0] = x; p2_ws[t * 3 + 1] = y; p2_ws[t * 3 + 2] = z;
    p2_out[t * 3 + 0] = x; p2_out[t * 3 + 1] = y; p2_out[t * 3 + 2] = z;
}

// ---------------------------------------------------------------------------
// kNN: whole batch point cloud staged in LDS (48KB), one query per thread,
// top-16 smallest kept via max-slot replacement fully in registers.
// ---------------------------------------------------------------------------
__global__ __launch_bounds__(256) void knn_kernel(const float* __restrict__ p1,
                                                  const float* __restrict__ p2,
                                                  int* __restrict__ idx_knn) {
    __shared__ float s_pts[NN_ * 3];
    const int b = blockIdx.x >> 2;
    const int m = ((blockIdx.x & 3) << 8) + threadIdx.x;
    for (int e = threadIdx.x; e < NN_ * 3; e += 256) s_pts[e] = p1[b * NN_ * 3 + e];
    __syncthreads();
    const int q = b * MM_ + m;
    const float qx = p2[q * 3 + 0], qy = p2[q * 3 + 1], qz = p2[q * 3 + 2];
    float bd[KNB_]; int bi[KNB_];
#pragma unroll
    for (int i = 0; i < KNB_; ++i) { bd[i] = 3.4e38f; bi[i] = 0; }
    float maxd = 3.4e38f; int maxs = 0;
    for (int j = 0; j < NN_; ++j) {
        float x = s_pts[j * 3 + 0], y = s_pts[j * 3 + 1], z = s_pts[j * 3 + 2];
        bool zero = (x == 0.f) && (y == 0.f) && (z == 0.f);   // ref zero-mask
        float dx = x - qx, dy = y - qy, dz = z - qz;
        float d = dx * dx + dy * dy + dz * dz;
        if (!zero && d < maxd) {
#pragma unroll
            for (int i = 0; i < KNB_; ++i)
                if (i == maxs) { bd[i] = d; bi[i] = j; }      // keep in registers
            maxd = bd[0]; maxs = 0;
#pragma unroll
            for (int i = 1; i < KNB_; ++i)
                if (bd[i] > maxd) { maxd = bd[i]; maxs = i; }
        }
    }
#pragma unroll
    for (int i = 0; i < KNB_; ++i) idx_knn[q * KNB_ + i] = bi[i];
}

// ---------------------------------------------------------------------------
// Fused gather + GEMM1(+BN1+ReLU) + GEMM2(+BN2+ReLU+maxpool) using f32 WMMA.
// MODE 0: stats of h1.  MODE 1: stats of h2.  MODE 2: final output.
// One 16x16 output tile per wave; 8 waves cover all 128 channels of one
// (b,m) group; GROUPS_ groups processed per block.
// ---------------------------------------------------------------------------
template <int MODE>
__global__ __launch_bounds__(256) void mlp_kernel(
    const float* __restrict__ p1, const float* __restrict__ x1,
    const int*   __restrict__ idx_knn, const float* __restrict__ p2,
    const float* __restrict__ w1, const float* __restrict__ w2,
    const float* __restrict__ bn1, const float* __restrict__ bn2,
    float* __restrict__ stats_part, float* __restrict__ out_feat) {

    __shared__ float s_feat[16 * K1P_];
    __shared__ float s_h1[16 * 132];
    __shared__ int   s_idx[16];
    __shared__ float s_p2[3];

    const int tid  = threadIdx.x;
    const int wv   = tid >> 5;
    const int lane = tid & 31;
    const int nn   = lane & 15;
    const bool hi  = lane >= 16;
    const int koff = hi ? 2 : 0;
    const int ch   = wv * 16 + nn;     // channel handled by this lane's column

    float gsum = 0.f, gsq = 0.f;

    for (int g = 0; g < GROUPS_; ++g) {
        const int gid = blockIdx.x * GROUPS_ + g;
        const int b = gid >> 10;
        const int m = gid & (MM_ - 1);

        if (tid < 16) s_idx[tid] = idx_knn[gid * KNB_ + tid];
        if (tid >= 16 && tid < 19) s_p2[tid - 16] = p2[gid * 3 + (tid - 16)];
        __syncthreads();

        // Gather feats tile: 16 neighbors x (rel_xyz[3] | x1[64] | pad0)
        for (int e = tid; e < 16 * K1P_; e += 256) {
            int r = e / K1P_, c = e - r * K1P_;
            int j = s_idx[r];
            float v;
            if (c < 3)        v = p1[(b * NN_ + j) * 3 + c] - s_p2[c];
            else if (c < K1_) v = x1[(b * CIN_ + (c - 3)) * NN_ + j];
            else              v = 0.f;
            s_feat[e] = v;
        }
        __syncthreads();

        // ---- GEMM1: 16x68 (A) x 68x16 (B=w1^T tile), K-step 4 f32 WMMA ----
        v8f acc = {};
        for (int k0 = 0; k0 < K1P_; k0 += 4) {
            int ka = k0 + koff;
            v2f a; a.x = s_feat[nn * K1P_ + ka];
                   a.y = s_feat[nn * K1P_ + ka + 1];
            v2f bb;
            bb.x = (ka     < K1_) ? w1[ch * K1_ + ka]     : 0.f;
            bb.y = (ka + 1 < K1_) ? w1[ch * K1_ + ka + 1] : 0.f;
            acc = __builtin_amdgcn_wmma_f32_16x16x4_f32(
                false, a, false, bb, (short)0, acc, false, false);
        }

        if (MODE == 0) {
            float s = 0.f, sq = 0.f;
#pragma unroll
            for (int i = 0; i < 8; ++i) { float v = acc[i]; s += v; sq += v * v; }
            gsum += s; gsq += sq;
            __syncthreads();
            continue;
        }

        // ---- BN1 + ReLU -> LDS (padded stride 132 to dodge bank conflicts)
        {
            float sc = bn1[ch], sh = bn1[128 + ch];
#pragma unroll
            for (int i = 0; i < 8; ++i) {
                float h = fmaxf(fmaf(acc[i], sc, sh), 0.f);
                int row = i + (hi ? 8 : 0);
                s_h1[row * 132 + ch] = h;
            }
        }
        __syncthreads();

        // ---- GEMM2: 16x128 x 128x16, 32 f32 WMMA steps ----
        v8f acc2 = {};
        for (int k0 = 0; k0 < COUT_; k0 += 4) {
            int ka = k0 + koff;
            v2f a; a.x = s_h1[nn * 132 + ka];
                   a.y = s_h1[nn * 132 + ka + 1];
            v2f bb; bb.x = w2[ch * COUT_ + ka];
                    bb.y = w2[ch * COUT_ + ka + 1];
            acc2 = __builtin_amdgcn_wmma_f32_16x16x4_f32(
                false, a, false, bb, (short)0, acc2, false, false);
        }

        if (MODE == 1) {
            float s = 0.f, sq = 0.f;
#pragma unroll
            for (int i = 0; i < 8; ++i) { float v = acc2[i]; s += v; sq += v * v; }
            gsum += s; gsq += sq;
        } else {
            // BN2 + ReLU + max over the 16 neighbor rows of this tile
            float sc = bn2[ch], sh = bn2[128 + ch];
            float rm = -3.4e38f;
#pragma unroll
            for (int i = 0; i < 8; ++i) {
                float h = fmaxf(fmaf(acc2[i], sc, sh), 0.f);
                rm = fmaxf(rm, h);
            }
            rm = fmaxf(rm, __shfl_xor(rm, 16, 32));   // combine row halves
            if (lane < 16)
                out_feat[((size_t)b * COUT_ + ch) * MM_ + m] = rm;
        }
        __syncthreads();
    }

    if (MODE == 0 || MODE == 1) {
        float ps = gsum + __shfl_xor(gsum, 16, 32);
        float pq = gsq  + __shfl_xor(gsq, 16, 32);
        if (lane < 16) {
            stats_part[(size_t)blockIdx.x * 256 + ch]       = ps;
            stats_part[(size_t)blockIdx.x * 256 + 128 + ch] = pq;
        }
    }
}

// Deterministic reduction of per-block stats partials.
__global__ void stats_reduce_kernel(const float* __restrict__ part,
                                    float* __restrict__ stats) {
    int c = blockIdx.x * blockDim.x + threadIdx.x;
    if (c < 256) {
        float s = 0.f;
        for (int i = 0; i < NBLK_; ++i) s += part[(size_t)i * 256 + c];
        stats[c] = s;
    }
}

// scale = g * rsqrt(var + eps); shift = b - mean * scale
__global__ void bn_finalize_kernel(const float* __restrict__ stats,
                                   const float* __restrict__ gg,
                                   const float* __restrict__ bbeta,
                                   float* __restrict__ bn) {
    int c = threadIdx.x;
    if (c < 128) {
        const float inv_cnt = 1.0f / (float)CNT_;
        float mean = stats[c] * inv_cnt;
        float var  = stats[128 + c] * inv_cnt - mean * mean;
        float sc   = gg[c] * rsqrtf(var + EPS_);
        bn[c]       = sc;
        bn[128 + c] = bbeta[c] - mean * sc;
    }
}

// ---------------------------------------------------------------------------
extern "C" void kernel_launch(void* const* d_in, const int* in_sizes, int n_in,
                              void* d_out, int out_size, void* d_ws, size_t ws_size,
                              hipStream_t stream) {
    const float* p1 = (const float*)d_in[0];
    const float* x1 = (const float*)d_in[1];
    const float* w1 = (const float*)d_in[2];
    const float* g1 = (const float*)d_in[3];
    const float* b1 = (const float*)d_in[4];
    const float* w2 = (const float*)d_in[5];
    const float* g2 = (const float*)d_in[6];
    const float* b2 = (const float*)d_in[7];
    float* out = (float*)d_out;
    char*  ws  = (char*)d_ws;

    int*   idx_fps = (int*)(ws + 0);                 //  64 KB
    float* p2      = (float*)(ws + 65536);           // 192 KB
    int*   idx_knn = (int*)(ws + 262144);            //   1 MB
    float* stats1  = (float*)(ws + 1310720);         //   1 KB
    float* stats2  = (float*)(ws + 1311744);         //   1 KB
    float* bn1     = (float*)(ws + 1312768);         //   1 KB
    float* bn2     = (float*)(ws + 1313792);         //   1 KB
    float* partial = (float*)(ws + 1314816);         //   2 MB

    float* p2_out   = out;                 // [B, M, 3]
    float* feat_out = out + BB_ * MM_ * 3; // [B, Cout, M]

    fps_kernel<<<BB_, 256, 0, stream>>>(p1, idx_fps);
    gather_p2_kernel<<<(BB_ * MM_ + 255) / 256, 256, 0, stream>>>(p1, idx_fps, p2, p2_out);
    knn_kernel<<<BB_ * 4, 256, 0, stream>>>(p1, p2, idx_knn);

    mlp_kernel<0><<<NBLK_, 256, 0, stream>>>(p1, x1, idx_knn, p2, w1, w2, bn1, bn2, partial, feat_out);
    stats_reduce_kernel<<<1, 256, 0, stream>>>(partial, stats1);
    bn_finalize_kernel<<<1, 128, 0, stream>>>(stats1, g1, b1, bn1);

    mlp_kernel<1><<<NBLK_, 256, 0, stream>>>(p1, x1, idx_knn, p2, w1, w2, bn1, bn2, partial, feat_out);
    stats_reduce_kernel<<<1, 256, 0, stream>>>(partial, stats2);
    bn_finalize_kernel<<<1, 128, 0, stream>>>(stats2, g2, b2, bn2);

    mlp_kernel<2><<<NBLK_, 256, 0, stream>>>(p1, x1, idx_knn, p2, w1, w2, bn1, bn2, partial, feat_out);
}